// A_GCN_72507637891556
// MI455X (gfx1250) — compile-verified
//
#include <hip/hip_runtime.h>
#include <hip/hip_bf16.h>

#define D_FEAT 64

typedef float v2f __attribute__((ext_vector_type(2)));
typedef float v8f __attribute__((ext_vector_type(8)));

// ---------------------------------------------------------------------------
// Utility: zero a buffer of 32-bit words
// ---------------------------------------------------------------------------
__global__ void gcn_zero_u32(unsigned* __restrict__ p, long long n) {
  long long i = (long long)blockIdx.x * blockDim.x + threadIdx.x;
  if (i < n) p[i] = 0u;
}

// ---------------------------------------------------------------------------
// Degree counting: integer atomics into (aliased) norm buffers
// ---------------------------------------------------------------------------
__global__ void gcn_degree(const int* __restrict__ src, const int* __restrict__ dst,
                           unsigned* __restrict__ outdeg, unsigned* __restrict__ indeg,
                           int n_edges) {
  int e = blockIdx.x * blockDim.x + threadIdx.x;
  if (e < n_edges) {
    atomicAdd(&outdeg[src[e]], 1u);
    atomicAdd(&indeg[dst[e]], 1u);
  }
}

// In-place: uint degree -> float rsqrt(max(deg,1))
__global__ void gcn_norms(float* __restrict__ outn, float* __restrict__ inn, int n) {
  int i = blockIdx.x * blockDim.x + threadIdx.x;
  if (i < n) {
    unsigned od = ((const unsigned*)outn)[i];
    unsigned id_ = ((const unsigned*)inn)[i];
    outn[i] = rsqrtf((float)(od > 1u ? od : 1u));
    inn[i]  = rsqrtf((float)(id_ > 1u ? id_ : 1u));
  }
}

// ---------------------------------------------------------------------------
// Edge scatter: agg[dst] += x[src] * out_norm[src]
// One thread per (edge, feature). 64 consecutive threads share one edge, so
// the x[src] gather is a coalesced 256B burst and src/dst are broadcast loads.
// ---------------------------------------------------------------------------
__global__ void gcn_scatter(const float* __restrict__ x,
                            const float* __restrict__ out_norm,
                            const int* __restrict__ src, const int* __restrict__ dst,
                            float* __restrict__ agg, int n_edges) {
  long long t = (long long)blockIdx.x * blockDim.x + threadIdx.x;
  int e = (int)(t >> 6);
  int f = (int)(t & 63);
  if (e < n_edges) {
    int s = src[e];
    int d = dst[e];
    float v = x[(size_t)s * D_FEAT + f] * out_norm[s];
    atomicAdd(&agg[(size_t)d * D_FEAT + f], v);
  }
}

// ---------------------------------------------------------------------------
// Dense projection via fp32 WMMA:
//   out = elu( (agg * in_norm[:,None]) @ W + b )
// Grid: one block per 16 rows, 4 waves per block, each wave owns a 16-col tile.
// V_WMMA_F32_16X16X4_F32, K=64 -> 16 WMMA steps per tile.
// EXEC is all-ones through the WMMA loop (no divergence before the stores).
// ---------------------------------------------------------------------------
__global__ __launch_bounds__(128) void gcn_gemm_elu(
    const float* __restrict__ agg, const float* __restrict__ in_norm,
    const float* __restrict__ W,     // 64x64 row-major (K x N)
    const float* __restrict__ bias,  // 64
    float* __restrict__ out, int n_nodes) {
  const int lane = threadIdx.x & 31;
  const int wave = threadIdx.x >> 5;  // 0..3 -> column tile
  const int row0 = blockIdx.x * 16;
  const int n0   = wave * 16;
  const int half = lane >> 4;   // 0: K pair {k0,k0+1}; 1: K pair {k0+2,k0+3}
  const int l15  = lane & 15;

  // A-fragment source row (clamped so every lane has a valid address; the
  // scale select keeps out-of-range rows mathematically zero without
  // perturbing EXEC).
  const int arow  = row0 + l15;
  const int arowc = arow < n_nodes ? arow : (n_nodes - 1);
  const float scale = (arow < n_nodes) ? in_norm[arowc] : 0.0f;
  const float* __restrict__ aptr = agg + (size_t)arowc * D_FEAT + 2 * half;
  // B-fragment: lane holds W[k0 + 2*half + j][n0 + l15], j = 0,1
  const float* __restrict__ bptr = W + (size_t)(2 * half) * D_FEAT + n0 + l15;

  v8f c = {};
#pragma unroll
  for (int k0 = 0; k0 < D_FEAT; k0 += 4) {
    v2f a;
    a.x = aptr[k0] * scale;
    a.y = aptr[k0 + 1] * scale;
    v2f b;
    b.x = bptr[(size_t)k0 * D_FEAT];
    b.y = bptr[(size_t)(k0 + 1) * D_FEAT];
    // (neg_a, A, neg_b, B, c_mod, C, reuse_a, reuse_b)
    c = __builtin_amdgcn_wmma_f32_16x16x4_f32(false, a, false, b,
                                              (short)0, c, false, false);
  }

  const float bv = bias[n0 + l15];
#pragma unroll
  for (int v = 0; v < 8; ++v) {
    int r = row0 + v + 8 * half;  // C/D layout: vgpr v -> M=v (lanes<16) / v+8
    if (r < n_nodes) {
      float xv = c[v] + bv;
      float yv = xv > 0.0f ? xv : expm1f(xv);  // jax.nn.elu
      out[(size_t)r * D_FEAT + n0 + l15] = yv;
    }
  }
}

// ---------------------------------------------------------------------------
// Host-side orchestration
// ---------------------------------------------------------------------------
extern "C" void kernel_launch(void* const* d_in, const int* in_sizes, int n_in,
                              void* d_out, int out_size, void* d_ws, size_t ws_size,
                              hipStream_t stream) {
  const float* feat = (const float*)d_in[0];
  const float* Ws[3] = {(const float*)d_in[1], (const float*)d_in[3], (const float*)d_in[5]};
  const float* bs[3] = {(const float*)d_in[2], (const float*)d_in[4], (const float*)d_in[6]};
  const int* src = (const int*)d_in[7];
  const int* dst = (const int*)d_in[8];

  const int n_nodes = in_sizes[0] / D_FEAT;
  const int n_edges = in_sizes[7];

  // Workspace layout
  float* out_norm = (float*)d_ws;                         // n_nodes
  float* in_norm  = out_norm + n_nodes;                   // n_nodes
  float* agg      = in_norm + n_nodes;                    // n_nodes * 64
  float* hbuf     = agg + (size_t)n_nodes * D_FEAT;       // n_nodes * 64
  float* outp     = (float*)d_out;

  const int TPB = 256;

  // Degrees -> norms (degree counters alias the norm buffers as uint32)
  {
    long long n2 = 2LL * n_nodes;
    gcn_zero_u32<<<(unsigned)((n2 + TPB - 1) / TPB), TPB, 0, stream>>>(
        (unsigned*)out_norm, n2);
    gcn_degree<<<(unsigned)((n_edges + TPB - 1) / TPB), TPB, 0, stream>>>(
        src, dst, (unsigned*)out_norm, (unsigned*)in_norm, n_edges);
    gcn_norms<<<(unsigned)((n_nodes + TPB - 1) / TPB), TPB, 0, stream>>>(
        out_norm, in_norm, n_nodes);
  }

  const float* x = feat;
  for (int l = 0; l < 3; ++l) {
    // agg = 0
    long long nagg = (long long)n_nodes * D_FEAT;
    gcn_zero_u32<<<(unsigned)((nagg + TPB - 1) / TPB), TPB, 0, stream>>>(
        (unsigned*)agg, nagg);
    // agg[dst] += x[src] * out_norm[src]
    long long nthreads = (long long)n_edges * D_FEAT;
    gcn_scatter<<<(unsigned)((nthreads + TPB - 1) / TPB), TPB, 0, stream>>>(
        x, out_norm, src, dst, agg, n_edges);
    // y = elu((agg * in_norm) @ W + b)
    float* y = (l == 2) ? outp : hbuf;
    unsigned gblocks = (unsigned)((n_nodes + 15) / 16);
    gcn_gemm_elu<<<gblocks, 128, 0, stream>>>(agg, in_norm, Ws[l], bs[l], y, n_nodes);
    x = y;
  }
}